// Splatting_38783554682908
// MI455X (gfx1250) — compile-verified
//
#include <hip/hip_runtime.h>
#include <cstdint>

// Problem constants (from reference): frame [16,8,512,512] f32, flow [16,2,512,512] f32
#define B_  16
#define C_  8
#define H_  512
#define W_  512

#define TILE     32          // tile edge handled per workgroup
#define RAD      4           // halo radius captured in LDS (flow ~ N(0,1))
#define PAD      (TILE + 2*RAD)   // 40
#define NTHREADS 256

#if __has_builtin(__builtin_amdgcn_tensor_load_to_lds) && __has_builtin(__builtin_amdgcn_s_wait_tensorcnt)
#define USE_TDM 1
#endif

typedef unsigned int v4u __attribute__((ext_vector_type(4)));
typedef int          v8i __attribute__((ext_vector_type(8)));
typedef int          v4i __attribute__((ext_vector_type(4)));

__global__ __launch_bounds__(NTHREADS) void splat_kernel(
    const float* __restrict__ frame,
    const float* __restrict__ flow,
    float* __restrict__ out)
{
    // LDS: per-tile accumulator (8 ch x 40 x 40) + staged flow tile (2 x 32 x 32)
    __shared__ float acc[C_ * PAD * PAD];        // 51200 B
    __shared__ float ftile[2 * TILE * TILE];     //  8192 B

    const int tx0 = blockIdx.x * TILE;
    const int ty0 = blockIdx.y * TILE;
    const int b   = blockIdx.z;
    const int tid = threadIdx.x;
    const size_t HW = (size_t)H_ * W_;

    const float* flow_b  = flow  + ((size_t)b * 2)  * HW + (size_t)ty0 * W_ + tx0;
    const float* frame_b = frame + ((size_t)b * C_) * HW;
    float*       out_b   = out   + ((size_t)b * C_) * HW;

#ifdef USE_TDM
    // Tensor Data Mover: async-load both flow planes' 32x32 tiles into LDS.
    // D# per CDNA5 ISA (2D, groups 2/3 zero).
    if (tid == 0) {
        const uint32_t ldsBase = (uint32_t)(uintptr_t)(void*)ftile;
        #pragma unroll
        for (int plane = 0; plane < 2; ++plane) {
            const uint64_t ga = (uint64_t)(uintptr_t)(flow_b + (size_t)plane * HW);
            v4u g0; v8i g1; v4i g2; v4i g3; v8i g4;
            g0[0] = 1u;                                            // count=1, user mode
            g0[1] = ldsBase + (uint32_t)plane * (TILE * TILE * 4); // lds_addr (bytes)
            g0[2] = (uint32_t)ga;                                  // global_addr[31:0]
            g0[3] = (uint32_t)((ga >> 32) & 0x01FFFFFFu)           // global_addr[56:32]
                  | (2u << 30);                                    // type=2 ("image")
            g1[0] = (int)(2u << 16);                // data_size=2 -> 4 bytes
            g1[1] = (int)((uint32_t)W_ << 16);      // tensor_dim0 (low16 @ bit48)
            g1[2] = (int)((uint32_t)H_ << 16);      // tensor_dim1 (low16 @ bit80)
            g1[3] = (int)((uint32_t)TILE << 16);    // tile_dim0 @ bits[127:112]
            g1[4] = TILE;                           // tile_dim1 @ bits[143:128]
            g1[5] = W_;                             // tensor_dim0_stride (low32)
            g1[6] = 0;
            g1[7] = 0;
            g2[0] = g2[1] = g2[2] = g2[3] = 0;
            g3[0] = g3[1] = g3[2] = g3[3] = 0;
            g4[0] = g4[1] = g4[2] = g4[3] = g4[4] = g4[5] = g4[6] = g4[7] = 0;
            // clang-23 / therock-10.0 arity: 6 args (extra int32x8 before cpol)
            __builtin_amdgcn_tensor_load_to_lds(g0, g1, g2, g3, g4, 0);
        }
    }
#else
    // Fallback: coalesced non-temporal loads of the flow tile into LDS.
    for (int i = tid; i < 2 * TILE * TILE; i += NTHREADS) {
        const int plane = i >> 10;
        const int r     = (i >> 5) & (TILE - 1);
        const int col   = i & (TILE - 1);
        ftile[i] = __builtin_nontemporal_load(
            flow_b + (size_t)plane * HW + (size_t)r * W_ + col);
    }
#endif

    // Zero the LDS accumulator while the TDM transfer is in flight.
    for (int i = tid; i < C_ * PAD * PAD; i += NTHREADS) acc[i] = 0.0f;

#ifdef USE_TDM
    __builtin_amdgcn_s_wait_tensorcnt(0);   // fence TDM before LDS reads
#endif
    __syncthreads();

    // ---- Splat phase: each thread handles 4 pixels of the 32x32 tile ----
    #pragma unroll
    for (int k = 0; k < (TILE * TILE) / NTHREADS; ++k) {
        const int p  = tid + k * NTHREADS;
        const int lx = p & (TILE - 1);
        const int ly = p >> 5;
        const int x  = tx0 + lx;
        const int y  = ty0 + ly;

        const float dx = ftile[ly * TILE + lx];
        const float dy = ftile[TILE * TILE + ly * TILE + lx];

        const float txf = (float)x + dx;
        const float tyf = (float)y + dy;
        const float x0f = floorf(txf);
        const float y0f = floorf(tyf);
        const float ax  = txf - x0f;
        const float ay  = tyf - y0f;
        const int   xi  = (int)x0f;
        const int   yi  = (int)y0f;

        const float w00 = (1.0f - ax) * (1.0f - ay);
        const float w10 = ax * (1.0f - ay);
        const float w01 = (1.0f - ax) * ay;
        const float w11 = ax * ay;

        const int plx = xi - (tx0 - RAD);
        const int ply = yi - (ty0 - RAD);
        const size_t src = (size_t)y * W_ + x;

        if ((unsigned)plx < (unsigned)(PAD - 1) && (unsigned)ply < (unsigned)(PAD - 1)) {
            // Fast path (~100% of pixels): all 4 corners land in the LDS tile.
            const int base = ply * PAD + plx;
            #pragma unroll
            for (int c = 0; c < C_; ++c) {
                const float f = __builtin_nontemporal_load(frame_b + (size_t)c * HW + src);
                float* a = &acc[c * (PAD * PAD) + base];
                atomicAdd(a,           f * w00);   // ds_add_f32
                atomicAdd(a + 1,       f * w10);
                atomicAdd(a + PAD,     f * w01);
                atomicAdd(a + PAD + 1, f * w11);
            }
        } else {
            // Rare outlier path: per-corner LDS / direct-global / drop.
            const int   cx[4] = { xi, xi + 1, xi,     xi + 1 };
            const int   cy[4] = { yi, yi,     yi + 1, yi + 1 };
            const float cw[4] = { w00, w10, w01, w11 };
            #pragma unroll
            for (int c = 0; c < C_; ++c) {
                const float f = __builtin_nontemporal_load(frame_b + (size_t)c * HW + src);
                #pragma unroll
                for (int q = 0; q < 4; ++q) {
                    const int lxp = cx[q] - (tx0 - RAD);
                    const int lyp = cy[q] - (ty0 - RAD);
                    if ((unsigned)lxp < (unsigned)PAD && (unsigned)lyp < (unsigned)PAD) {
                        atomicAdd(&acc[c * (PAD * PAD) + lyp * PAD + lxp], f * cw[q]);
                    } else if ((unsigned)cx[q] < (unsigned)W_ && (unsigned)cy[q] < (unsigned)H_) {
                        unsafeAtomicAdd(out_b + (size_t)c * HW + (size_t)cy[q] * W_ + cx[q],
                                        f * cw[q]);  // global_atomic_add_f32
                    } // else: out of bounds -> dropped (matches reference)
                }
            }
        }
    }

    __syncthreads();

    // ---- Flush phase: LDS accumulator -> global with f32 atomics ----
    for (int i = tid; i < C_ * PAD * PAD; i += NTHREADS) {
        const float v = acc[i];
        if (v != 0.0f) {                       // skip untouched halo cells
            const int c   = i / (PAD * PAD);
            const int r   = i - c * (PAD * PAD);
            const int lyp = r / PAD;
            const int lxp = r - lyp * PAD;
            const int gx  = tx0 - RAD + lxp;
            const int gy  = ty0 - RAD + lyp;
            if ((unsigned)gx < (unsigned)W_ && (unsigned)gy < (unsigned)H_) {
                unsafeAtomicAdd(out_b + (size_t)c * HW + (size_t)gy * W_ + gx, v);
            }
        }
    }
}

extern "C" void kernel_launch(void* const* d_in, const int* in_sizes, int n_in,
                              void* d_out, int out_size, void* d_ws, size_t ws_size,
                              hipStream_t stream)
{
    (void)in_sizes; (void)n_in; (void)d_ws; (void)ws_size;
    const float* frame = (const float*)d_in[0];
    const float* flow  = (const float*)d_in[1];
    float*       out   = (float*)d_out;

    // Output must start at zero every call (harness poisons it).
    (void)hipMemsetAsync(d_out, 0, (size_t)out_size * sizeof(float), stream);

    dim3 grid(W_ / TILE, H_ / TILE, B_);   // 16 x 16 x 16 = 4096 workgroups
    splat_kernel<<<grid, NTHREADS, 0, stream>>>(frame, flow, out);
}